// RipsModule_36541581754617
// MI455X (gfx1250) — compile-verified
//
#include <hip/hip_runtime.h>
#include <hip/hip_bf16.h>

typedef __attribute__((ext_vector_type(2))) float v2f;
typedef __attribute__((ext_vector_type(8))) float v8f;

#define D_DIM 64

// ---------------------------------------------------------------------------
// Kernel 1: squared row norms.  One wave (32 lanes) per row of X[N, 64].
// Lane l loads X[row][2l], X[row][2l+1] (fully coalesced b64 loads), squares,
// butterfly-reduces across the wave, lane 0 writes norms[row].
// ---------------------------------------------------------------------------
__global__ __launch_bounds__(256) void rips_norms_kernel(const float* __restrict__ X,
                                                         float* __restrict__ norms,
                                                         int N) {
    const int wid  = (blockIdx.x * blockDim.x + threadIdx.x) >> 5;
    const int lane = threadIdx.x & 31;
    if (wid >= N) return;

    const float* row = X + (size_t)wid * D_DIM + 2 * lane;
    v2f v = *(const v2f*)row;
    float s = v.x * v.x + v.y * v.y;

    #pragma unroll
    for (int off = 16; off > 0; off >>= 1)
        s += __shfl_xor(s, off, 32);

    if (lane == 0) norms[wid] = s;
}

// ---------------------------------------------------------------------------
// Kernel 2: tiled fp32 WMMA GEMM  G = X · X^T  + distance finalize.
//
// Per ISA 16x16x4 f32 layouts (wave32):
//   A (16x4):  lane holds M = lane&15; VGPR0/1 = K = 2*(lane>>4) + {0,1}
//   B (4x16):  lane holds N = lane&15; VGPR0/1 = K = 2*(lane>>4) + {0,1}
//   C (16x16): VGPR r = row (r + 8*(lane>>4)), col = lane&15
// => both A and B fragments are one global_load_b64 per k-step from X,
//    at byte offset  row*256 + (4s + 2*(lane>>4))*4.
//
// Block = 256 threads = 8 waves arranged 4 (rows) x 2 (cols); each wave
// computes a 16x64 output strip (A fragment reused over 4 j-tiles).
// Block output tile: 64 x 128.  Grid: (N/128, N/64).
// ---------------------------------------------------------------------------
__global__ __launch_bounds__(256) void rips_dist_kernel(const float* __restrict__ X,
                                                        const float* __restrict__ norms,
                                                        float* __restrict__ out,
                                                        int N) {
    const int lane = threadIdx.x & 31;
    const int w    = threadIdx.x >> 5;     // wave id 0..7
    const int m    = lane & 15;
    const int kh   = lane >> 4;            // 0 or 1: which K-half this lane holds

    const int i0 = blockIdx.y * 64  + (w >> 1) * 16;
    const int j0 = blockIdx.x * 128 + (w & 1)  * 64;

    // ---- A fragments: 16 k-steps, reused across 4 j-tiles ----
    v2f a[16];
    const float* Ap = X + (size_t)(i0 + m) * D_DIM + 2 * kh;
    #pragma unroll
    for (int s = 0; s < 16; ++s)
        a[s] = *(const v2f*)(Ap + 4 * s);

    // ---- row norms for this wave's 16 output rows (split by lane half) ----
    float ni[8];
    #pragma unroll
    for (int r = 0; r < 8; ++r)
        ni[r] = norms[i0 + 8 * kh + r];

    #pragma unroll
    for (int t = 0; t < 4; ++t) {
        const int jt = j0 + 16 * t;

        // ---- B fragments for this j-tile ----
        v2f b[16];
        const float* Bp = X + (size_t)(jt + m) * D_DIM + 2 * kh;
        #pragma unroll
        for (int s = 0; s < 16; ++s)
            b[s] = *(const v2f*)(Bp + 4 * s);

        // ---- K-loop: 16 x V_WMMA_F32_16X16X4_F32 ----
        v8f c = {0.f, 0.f, 0.f, 0.f, 0.f, 0.f, 0.f, 0.f};
        #pragma unroll
        for (int s = 0; s < 16; ++s)
            c = __builtin_amdgcn_wmma_f32_16x16x4_f32(
                    /*neg_a=*/false, a[s],
                    /*neg_b=*/false, b[s],
                    /*c_mod=*/(short)0, c,
                    /*reuse_a=*/false, /*reuse_b=*/false);

        // ---- finalize: d = sqrt(max(ni + nj - 2*G, 0)), NT streaming store ----
        const float nj = norms[jt + m];
        float* orow = out + (size_t)(i0 + 8 * kh) * N + (jt + m);
        #pragma unroll
        for (int r = 0; r < 8; ++r) {
            float sq = ni[r] + nj - 2.0f * c[r];
            sq = fmaxf(sq, 0.0f);
            __builtin_nontemporal_store(__builtin_sqrtf(sq), orow + (size_t)r * N);
        }
    }
}

// ---------------------------------------------------------------------------
// Launch: kernel1 writes norms into d_ws (N floats), kernel2 consumes them.
// Deterministic; all work on `stream`; no mallocs/syncs (graph-capture safe).
// ---------------------------------------------------------------------------
extern "C" void kernel_launch(void* const* d_in, const int* in_sizes, int n_in,
                              void* d_out, int out_size, void* d_ws, size_t ws_size,
                              hipStream_t stream) {
    const float* X   = (const float*)d_in[0];
    float* out       = (float*)d_out;
    float* norms     = (float*)d_ws;

    const int N = in_sizes[0] / D_DIM;   // 8192 for the reference setup

    // Kernel 1: one wave per row, 8 waves per block.
    const int rows_per_block = 256 / 32;
    const int nblocks1 = (N + rows_per_block - 1) / rows_per_block;
    rips_norms_kernel<<<nblocks1, 256, 0, stream>>>(X, norms, N);

    // Kernel 2: block tile 64 (rows) x 128 (cols).
    dim3 grid(N / 128, N / 64);
    rips_dist_kernel<<<grid, 256, 0, stream>>>(X, norms, out, N);
}